// BEiT_37340445672146
// MI455X (gfx1250) — compile-verified
//
#include <hip/hip_runtime.h>
#include <hip/hip_bf16.h>

typedef __attribute__((ext_vector_type(2))) float v2f;
typedef __attribute__((ext_vector_type(8))) float v8f;

#define B_      4
#define N_      1025
#define C_      1024
#define NH      16
#define HD      64
#define M_TOT   (B_ * N_)          // 4100
#define SCALE_Q 0.125f             // 64^-0.5
#define OLD_G   47                 // 2*24-1
#define NEW_G   63                 // 2*32-1
#define NEW_NUM 3972               // 63*63+3
#define MT_QKV  257                // ceil(4100/16)
#define OT_QKV  192                // 3072/16
#define OT_PRJ  64                 // 1024/16
#define QTILES  65                 // ceil(1025/16)
#define KTILES  65
#define QBLK    17                 // ceil(65/4) q-tile blocks per (b,h)

// --------------------------- async global->LDS ------------------------------
#if defined(__has_builtin)
#  if __has_builtin(__builtin_amdgcn_global_load_async_to_lds_b128)
#    define HAVE_ASYNC_LDS 1
#  endif
#endif
#ifndef HAVE_ASYNC_LDS
#  define HAVE_ASYNC_LDS 0
#endif

// prototype is (int4 AS1*, int4 AS3*, imm offset, imm cpol) per hipcc diagnostic
typedef int v4i_ __attribute__((vector_size(16)));
typedef __attribute__((address_space(1))) v4i_* gv4i_p;   // global int4*
typedef __attribute__((address_space(3))) v4i_* lv4i_p;   // LDS    int4*

__device__ __forceinline__ void async_cp16B(const float* g, float* l) {
#if HAVE_ASYNC_LDS
    __builtin_amdgcn_global_load_async_to_lds_b128((gv4i_p)g, (lv4i_p)l, 0, 0);
#else
    *(float4*)l = *(const float4*)g;            // synchronous fallback
#endif
}
__device__ __forceinline__ void wait_async_le4() {
#if HAVE_ASYNC_LDS
    asm volatile("s_wait_asynccnt 4" ::: "memory");
#endif
}
__device__ __forceinline__ void wait_async_le0() {
#if HAVE_ASYNC_LDS
    asm volatile("s_wait_asynccnt 0" ::: "memory");
#endif
}

// ---------------------------------------------------------------------------
// 1) Bilinear-resize the (47x47,16) grid part of the rel-pos table to (63x63,16)
// ---------------------------------------------------------------------------
__global__ void k_resize(const float* __restrict__ tbl, float* __restrict__ ntab) {
    int idx = blockIdx.x * blockDim.x + threadIdx.x;
    const int total = NEW_NUM * NH;
    if (idx >= total) return;
    int h   = idx % NH;
    int row = idx / NH;
    if (row >= NEW_G * NEW_G) {                 // 3 cls rows copied verbatim
        int c = row - NEW_G * NEW_G;
        ntab[idx] = tbl[(OLD_G * OLD_G + c) * NH + h];
        return;
    }
    int a = row / NEW_G, b = row % NEW_G;
    const float r = (float)OLD_G / (float)NEW_G;
    float fy = (a + 0.5f) * r - 0.5f;
    float fx = (b + 0.5f) * r - 0.5f;
    float yf = floorf(fy), xf = floorf(fx);
    float wy = fy - yf,    wx = fx - xf;
    int y0 = (int)yf, x0 = (int)xf;
    int y1 = y0 + 1,  x1 = x0 + 1;
    y0 = max(0, min(OLD_G - 1, y0)); y1 = max(0, min(OLD_G - 1, y1));
    x0 = max(0, min(OLD_G - 1, x0)); x1 = max(0, min(OLD_G - 1, x1));
    float v00 = tbl[(y0 * OLD_G + x0) * NH + h];
    float v01 = tbl[(y0 * OLD_G + x1) * NH + h];
    float v10 = tbl[(y1 * OLD_G + x0) * NH + h];
    float v11 = tbl[(y1 * OLD_G + x1) * NH + h];
    ntab[idx] = v00 * (1.f - wy) * (1.f - wx) + v01 * (1.f - wy) * wx +
                v10 * wy * (1.f - wx)         + v11 * wy * wx;
}

// ---------------------------------------------------------------------------
// 2) QKV GEMM: one wave = 16x16 tile, 256 x v_wmma_f32_16x16x4_f32
// ---------------------------------------------------------------------------
__global__ void k_qkv(const float* __restrict__ x, const float* __restrict__ w,
                      const float* __restrict__ qb, const float* __restrict__ kb,
                      const float* __restrict__ vb,
                      float* __restrict__ q, float* __restrict__ kmat,
                      float* __restrict__ v) {
    int wave = blockIdx.x * (blockDim.x >> 5) + (threadIdx.x >> 5);
    int lane = threadIdx.x & 31;
    if (wave >= MT_QKV * OT_QKV) return;
    int mt = wave % MT_QKV, ot = wave / MT_QKV;
    int M0 = mt * 16, O0 = ot * 16;
    int half = lane >> 4, l16 = lane & 15;

    int arow = min(M0 + l16, M_TOT - 1);
    const float* ap = x + (size_t)arow * C_ + half * 2;
    const float* bp = w + (size_t)(O0 + l16) * C_ + half * 2;

    v8f acc = {};
    for (int k0 = 0; k0 < C_; k0 += 4) {
        if ((k0 & 63) == 0 && k0 + 260 <= C_) {
            __builtin_prefetch(ap + k0 + 256, 0, 3);
            __builtin_prefetch(bp + k0 + 256, 0, 3);
        }
        v2f a = *(const v2f*)(ap + k0);
        v2f b = *(const v2f*)(bp + k0);
        acc = __builtin_amdgcn_wmma_f32_16x16x4_f32(false, a, false, b,
                                                    (short)0, acc, false, false);
    }

    int o     = O0 + l16;             // [0,3072)
    int which = o / C_;               // 0=q 1=k 2=v (uniform per tile)
    int cc    = o % C_;
    int h     = cc / HD;
    int d     = cc % HD;
    float bias = (which == 0) ? qb[cc] : (which == 1) ? kb[cc] : vb[cc];
    float mul  = (which == 0) ? SCALE_Q : 1.0f;
    float* dst = (which == 0) ? q : (which == 1) ? kmat : v;

#pragma unroll
    for (int r = 0; r < 8; r++) {
        int m = M0 + r + 8 * half;
        if (m >= M_TOT) continue;
        int bb = m / N_, nn = m % N_;
        dst[(((size_t)bb * NH + h) * N_ + nn) * HD + d] = (acc[r] + bias) * mul;
    }
}

// ---------------------------------------------------------------------------
// 3) Flash attention.  128 threads = 4 waves; each wave owns one 16-row q-tile
//    of the same (b,h).  K/V tiles are double-buffered in LDS via async
//    global->LDS copies (ASYNCcnt) shared by all 4 waves; softmax is online
//    (shfl_xor 16-lane reductions); P relaid D->A layout through LDS.
// ---------------------------------------------------------------------------
__global__ __launch_bounds__(128) void k_attn(
        const float* __restrict__ q, const float* __restrict__ k,
        const float* __restrict__ v, const float* __restrict__ ntab,
        float* __restrict__ ctx) {
    __shared__ float sK[2][16 * 64];
    __shared__ float sV[2][16 * 64];
    __shared__ float sP[4][16 * 16];

    int tid  = threadIdx.x;
    int widx = tid >> 5, lane = tid & 31;
    int half = lane >> 4, l16 = lane & 15;
    int bh   = blockIdx.x / QBLK;
    int qblk = blockIdx.x % QBLK;
    int b    = bh / NH, h = bh % NH;
    int qt_raw = qblk * 4 + widx;               // 0..67
    int qt     = min(qt_raw, QTILES - 1);
    int q0     = qt * 16;

    const float* qp = q + (((size_t)b * NH + h) * N_) * HD;
    const float* kp = k + (((size_t)b * NH + h) * N_) * HD;
    const float* vp = v + (((size_t)b * NH + h) * N_) * HD;

    // Q tile in A-layout: 16 chunks of K=4 over the 64-dim head
    v2f qa[16];
    int qrow_ld = min(q0 + l16, N_ - 1);
#pragma unroll
    for (int c = 0; c < 16; c++)
        qa[c] = *(const v2f*)(qp + (size_t)qrow_ld * HD + c * 4 + half * 2);

    v8f o[4] = {{}, {}, {}, {}};
    float mrow[8], lrow[8];
#pragma unroll
    for (int r = 0; r < 8; r++) { mrow[r] = -1e30f; lrow[r] = 0.0f; }

    // Prologue: stage tile 0 (each thread copies 2x16B for K and for V:
    // a 16x64 f32 tile is one contiguous 4KB block in the (B,H,N,64) layout)
    async_cp16B(kp + tid * 4,         &sK[0][tid * 4]);
    async_cp16B(kp + (tid + 128) * 4, &sK[0][(tid + 128) * 4]);
    async_cp16B(vp + tid * 4,         &sV[0][tid * 4]);
    async_cp16B(vp + (tid + 128) * 4, &sV[0][(tid + 128) * 4]);

    for (int kt = 0; kt < KTILES; kt++) {
        int cur = kt & 1, nxt = cur ^ 1;
        if (kt < KTILES - 1) {                  // stage tile kt+1
            const float* kn = kp + (size_t)(kt + 1) * 16 * HD;
            const float* vn = vp + (size_t)(kt + 1) * 16 * HD;
            async_cp16B(kn + tid * 4,         &sK[nxt][tid * 4]);
            async_cp16B(kn + (tid + 128) * 4, &sK[nxt][(tid + 128) * 4]);
            async_cp16B(vn + tid * 4,         &sV[nxt][tid * 4]);
            async_cp16B(vn + (tid + 128) * 4, &sV[nxt][(tid + 128) * 4]);
            wait_async_le4();                   // dones retire in order: tile kt done
        } else {
            wait_async_le0();
        }
        __syncthreads();                        // whole tile visible to all 4 waves

        const float* sKc = sK[cur];
        const float* sVc = sV[cur];
        int ki      = kt * 16 + l16;
        bool kvalid = (ki < N_);

        // ---- S = Q x K^T (16x16 tile), K from LDS ----
        v8f s = {};
#pragma unroll
        for (int c = 0; c < 16; c++) {
            v2f bm = *(const v2f*)(&sKc[l16 * HD + c * 4 + half * 2]);
            s = __builtin_amdgcn_wmma_f32_16x16x4_f32(false, qa[c], false, bm,
                                                      (short)0, s, false, false);
        }

        // ---- relative position bias + k-range mask ----
#pragma unroll
        for (int r = 0; r < 8; r++) {
            if (!kvalid) { s[r] = -1e30f; continue; }
            int qi = min(q0 + r + 8 * half, N_ - 1);
            int idx;
            if (qi == 0 && ki == 0)      idx = NEW_NUM - 1;
            else if (qi == 0)            idx = NEW_NUM - 3;
            else if (ki == 0)            idx = NEW_NUM - 2;
            else {
                int p  = qi - 1, pk = ki - 1;
                int dy = (p >> 5) - (pk >> 5) + 31;
                int dx = (p & 31) - (pk & 31) + 31;
                idx = dy * NEW_G + dx;
            }
            s[r] += ntab[idx * NH + h];
        }

        // ---- online softmax ----
        float alpha[8];
#pragma unroll
        for (int r = 0; r < 8; r++) {
            float m = s[r];
            m = fmaxf(m, __shfl_xor(m, 1, 32));
            m = fmaxf(m, __shfl_xor(m, 2, 32));
            m = fmaxf(m, __shfl_xor(m, 4, 32));
            m = fmaxf(m, __shfl_xor(m, 8, 32));
            float mn = fmaxf(mrow[r], m);
            alpha[r] = __expf(mrow[r] - mn);
            mrow[r]  = mn;
            float p  = __expf(s[r] - mn);
            s[r]     = p;
            float t  = p;
            t += __shfl_xor(t, 1, 32);
            t += __shfl_xor(t, 2, 32);
            t += __shfl_xor(t, 4, 32);
            t += __shfl_xor(t, 8, 32);
            lrow[r] = lrow[r] * alpha[r] + t;
        }

        // ---- rescale accumulators ----
#pragma unroll
        for (int vt = 0; vt < 4; vt++)
#pragma unroll
            for (int r = 0; r < 8; r++) o[vt][r] *= alpha[r];

        // ---- P: D-layout -> A-layout through per-wave LDS region ----
        // (same-wave LDS ops are in-order; region is private -> no barrier)
        float* myP = sP[widx];
#pragma unroll
        for (int r = 0; r < 8; r++)
            myP[(r + 8 * half) * 16 + l16] = s[r];
        v2f pa[4];
#pragma unroll
        for (int c = 0; c < 4; c++)
            pa[c] = *(const v2f*)(&myP[l16 * 16 + c * 4 + half * 2]);

        // ---- O += P x V, V from LDS ----
#pragma unroll
        for (int vt = 0; vt < 4; vt++) {
#pragma unroll
            for (int c = 0; c < 4; c++) {
                int kr = c * 4 + half * 2;
                v2f bm;
                bm[0] = sVc[kr * HD + vt * 16 + l16];
                bm[1] = sVc[(kr + 1) * HD + vt * 16 + l16];
                o[vt] = __builtin_amdgcn_wmma_f32_16x16x4_f32(false, pa[c], false, bm,
                                                              (short)0, o[vt], false, false);
            }
        }
        __syncthreads();   // nobody overwrites buf 'nxt' while a peer reads 'cur'
    }

    // ---- normalize, store to ctx (B, N, C) ----
    if (qt_raw < QTILES) {
        float* cp = ctx + ((size_t)b * N_) * C_ + h * HD;
#pragma unroll
        for (int r = 0; r < 8; r++) {
            int qi = q0 + r + 8 * half;
            if (qi >= N_) continue;
            float inv = 1.0f / lrow[r];
#pragma unroll
            for (int vt = 0; vt < 4; vt++)
                cp[(size_t)qi * C_ + vt * 16 + l16] = o[vt][r] * inv;
        }
    }
}

// ---------------------------------------------------------------------------
// 4) Output projection: (4100 x 1024) x (1024 x 1024)^T + bias -> d_out
// ---------------------------------------------------------------------------
__global__ void k_proj(const float* __restrict__ ctx, const float* __restrict__ w,
                       const float* __restrict__ pb, float* __restrict__ out) {
    int wave = blockIdx.x * (blockDim.x >> 5) + (threadIdx.x >> 5);
    int lane = threadIdx.x & 31;
    if (wave >= MT_QKV * OT_PRJ) return;
    int mt = wave % MT_QKV, ot = wave / MT_QKV;
    int M0 = mt * 16, O0 = ot * 16;
    int half = lane >> 4, l16 = lane & 15;

    int arow = min(M0 + l16, M_TOT - 1);
    const float* ap = ctx + (size_t)arow * C_ + half * 2;
    const float* bp = w + (size_t)(O0 + l16) * C_ + half * 2;

    v8f acc = {};
    for (int k0 = 0; k0 < C_; k0 += 4) {
        if ((k0 & 63) == 0 && k0 + 260 <= C_) {
            __builtin_prefetch(ap + k0 + 256, 0, 3);
            __builtin_prefetch(bp + k0 + 256, 0, 3);
        }
        v2f a = *(const v2f*)(ap + k0);
        v2f b = *(const v2f*)(bp + k0);
        acc = __builtin_amdgcn_wmma_f32_16x16x4_f32(false, a, false, b,
                                                    (short)0, acc, false, false);
    }
    float bias = pb[O0 + l16];
#pragma unroll
    for (int r = 0; r < 8; r++) {
        int m = M0 + r + 8 * half;
        if (m >= M_TOT) continue;
        out[(size_t)m * C_ + O0 + l16] = acc[r] + bias;
    }
}

// ---------------------------------------------------------------------------
extern "C" void kernel_launch(void* const* d_in, const int* in_sizes, int n_in,
                              void* d_out, int out_size, void* d_ws, size_t ws_size,
                              hipStream_t stream) {
    const float* x    = (const float*)d_in[0];
    const float* qkvw = (const float*)d_in[1];
    const float* qb   = (const float*)d_in[2];
    const float* kb   = (const float*)d_in[3];
    const float* vb   = (const float*)d_in[4];
    const float* pw   = (const float*)d_in[5];
    const float* pb   = (const float*)d_in[6];
    const float* tbl  = (const float*)d_in[7];
    float* out = (float*)d_out;
    float* ws  = (float*)d_ws;

    const size_t QKVSZ = (size_t)B_ * NH * N_ * HD;   // 4,198,400 floats
    float* q    = ws;
    float* kbuf = q + QKVSZ;
    float* vbuf = kbuf + QKVSZ;
    float* ctx  = vbuf + QKVSZ;
    float* ntab = ctx + (size_t)M_TOT * C_;

    {   // 1) resize rel-pos table
        int tot = NEW_NUM * NH;
        k_resize<<<(tot + 255) / 256, 256, 0, stream>>>(tbl, ntab);
    }
    {   // 2) qkv projection
        int waves = MT_QKV * OT_QKV;
        k_qkv<<<(waves + 3) / 4, 128, 0, stream>>>(x, qkvw, qb, kb, vb,
                                                   q, kbuf, vbuf);
    }
    {   // 3) flash attention (4 waves / block share async-staged K,V tiles)
        k_attn<<<B_ * NH * QBLK, 128, 0, stream>>>(q, kbuf, vbuf, ntab, ctx);
    }
    {   // 4) output projection
        int waves = MT_QKV * OT_PRJ;
        k_proj<<<(waves + 3) / 4, 128, 0, stream>>>(ctx, pw, pb, out);
    }
}